// DFM_26594437497357
// MI455X (gfx1250) — compile-verified
//
#include <hip/hip_runtime.h>
#include <math.h>

// Problem constants (match reference)
#define B_N   8
#define C_N   64
#define K_N   32
#define H_N   128
#define W_N   128
#define HW_N  (H_N * W_N)      // 16384
#define NEG_SLOPE 0.1f

// Stage-1 decomposition
#define NSPLIT  8                           // N-dimension split (parallelism)
#define NC      (HW_N / NSPLIT)             // 2048 n-values per workgroup
#define NCHUNK  64                          // n-values staged in LDS per step
#define CHUNKS  (NC / NCHUNK)               // 32 double-buffered steps

typedef __attribute__((ext_vector_type(2))) float v2f;
typedef __attribute__((ext_vector_type(8))) float v8f;
typedef __attribute__((vector_size(16)))    int   gv4i;   // matches builtin param

#define AS1 __attribute__((address_space(1)))
#define AS3 __attribute__((address_space(3)))

#if __has_builtin(__builtin_amdgcn_global_load_async_to_lds_b128) && \
    __has_builtin(__builtin_amdgcn_s_wait_asynccnt)
#define USE_ASYNC 1
#else
#define USE_ASYNC 0
#endif

__device__ __forceinline__ void async_ld_b128(float* lds, const float* g) {
#if USE_ASYNC
    __builtin_amdgcn_global_load_async_to_lds_b128(
        (AS1 gv4i*)g, (AS3 gv4i*)lds, 0, 0);
#else
    *(float4*)lds = *(const float4*)g;   // synchronous fallback
#endif
}

// ---------------------------------------------------------------------------
// Stage 1: partial[s,b,tile] = sum over n-range s of LR[b,i,n]*HR[b,j,n].
// 16 waves per workgroup (one per 16x16 tile of the 64x64 corr matrix);
// LR/HR chunks staged once into LDS via async b128 loads (double-buffered),
// WMMA fragments read from LDS with an XOR swizzle (conflict-free b64 reads).
//
// A layout (ISA 7.12.2, 32-bit A 16x4): lane l -> row m=l&15, half h=l>>4;
//   vgpr0 = K=2h, vgpr1 = K=2h+1.  B mirrored (col n=l&15).
// C/D layout: vgpr r, lane l -> M = r + 8*h, N = l&15.
// ---------------------------------------------------------------------------
__global__ __launch_bounds__(512)
void corr_stage_kernel(const float* __restrict__ LR,
                       const float* __restrict__ HR,
                       float* __restrict__ partial) {
    __shared__ float sA[2][C_N * NCHUNK];   // 32 KB
    __shared__ float sB[2][C_N * NCHUNK];   // 32 KB  (total 64 KB of 320 KB/WGP)

    const int s    = blockIdx.x;        // n-split index
    const int b    = blockIdx.y;        // batch
    const int t    = threadIdx.x;       // 0..511
    const int wv   = t >> 5;            // wave 0..15 -> tile (ti,tj)
    const int lane = t & 31;
    const int hl   = lane >> 4;
    const int lm   = lane & 15;
    const int ti   = wv >> 2;
    const int tj   = wv & 3;

    const size_t base = (size_t)b * C_N * HW_N + (size_t)s * NC;

    // Loader: 64 rows x 16 float4 = 1024 b128 per matrix; 512 threads -> 2 each.
    // LDS col is XOR-swizzled by 4*(row&15): keeps 16B alignment, and makes the
    // 16-lane b64 fragment reads land on all 64 banks with no conflicts.
    auto stage = [&](int c, int buf) {
        const int n0 = c * NCHUNK;
#pragma unroll
        for (int e0 = 0; e0 < 2; ++e0) {
            const int e   = t + e0 * 512;
            const int row = e >> 4;
            const int col = (e & 15) * 4;
            const int sc  = col ^ (4 * (row & 15));
            async_ld_b128(&sA[buf][row * NCHUNK + sc],
                          LR + base + (size_t)row * HW_N + n0 + col);
            async_ld_b128(&sB[buf][row * NCHUNK + sc],
                          HR + base + (size_t)row * HW_N + n0 + col);
        }
    };

    v8f acc = {};
    stage(0, 0);

    for (int c = 0; c < CHUNKS; ++c) {
        const int buf = c & 1;
#if USE_ASYNC
        if (c + 1 < CHUNKS) {
            stage(c + 1, buf ^ 1);
            __builtin_amdgcn_s_wait_asynccnt(4);  // chunk c landed (in-order)
        } else {
            __builtin_amdgcn_s_wait_asynccnt(0);
        }
#else
        if (c + 1 < CHUNKS) stage(c + 1, buf ^ 1);
#endif
        __syncthreads();   // all waves' async data visible before fragment reads

        const float* __restrict__ Ab = &sA[buf][(ti * 16 + lm) * NCHUNK];
        const float* __restrict__ Bb = &sB[buf][(tj * 16 + lm) * NCHUNK];
#pragma unroll
        for (int u = 0; u < NCHUNK / 4; ++u) {      // 16 WMMAs per chunk
            const int cn = (4 * u + 2 * hl) ^ (4 * lm);   // un-swizzle
            v2f a;  a.x  = Ab[cn]; a.y  = Ab[cn + 1];
            v2f bb; bb.x = Bb[cn]; bb.y = Bb[cn + 1];
            acc = __builtin_amdgcn_wmma_f32_16x16x4_f32(
                /*neg_a=*/false, a, /*neg_b=*/false, bb,
                /*c_mod=*/(short)0, acc, /*reuse_a=*/false, /*reuse_b=*/false);
        }
        __syncthreads();   // protect buf from being restaged next iteration
    }

    // Tile-contiguous partial store: P[s][b][wv][256]
    float* __restrict__ p = partial + (((size_t)s * B_N + b) * 16 + wv) * 256;
#pragma unroll
    for (int r = 0; r < 8; ++r) p[(r + 8 * hl) * 16 + lm] = acc[r];
}

// ---------------------------------------------------------------------------
// Stage 2: fold NSPLIT partials (fixed order -> deterministic), then stable
// descending top-32 indices: rank(e) = #{v' > v} + #{v'==v, earlier index},
// which matches jax.lax.top_k ordering. One wave per (b, c_lr) row.
// ---------------------------------------------------------------------------
__global__ __launch_bounds__(32)
void topk_kernel(const float* __restrict__ partial, int* __restrict__ idx) {
    __shared__ float vals[C_N];
    const int t  = threadIdx.x;
    const int bc = blockIdx.x;        // b*64 + i
    const int b  = bc >> 6;
    const int i  = bc & 63;
    const int ti = i >> 4;
    const int ir = i & 15;

#pragma unroll
    for (int e0 = 0; e0 < 2; ++e0) {
        const int j  = t + 32 * e0;
        const int tj = j >> 4;
        const int jr = j & 15;
        float v = 0.f;
#pragma unroll
        for (int s = 0; s < NSPLIT; ++s)
            v += partial[(((size_t)s * B_N + b) * 16 + (ti * 4 + tj)) * 256
                         + ir * 16 + jr];
        vals[j] = v;
    }
    __syncthreads();

    int* __restrict__ orow = idx + (size_t)bc * K_N;
#pragma unroll
    for (int e0 = 0; e0 < 2; ++e0) {
        const int e = t + 32 * e0;
        const float v = vals[e];
        int rank = 0;
#pragma unroll 8
        for (int j = 0; j < C_N; ++j) {
            const float u = vals[j];
            rank += (u > v) | ((u == v) & (j < e));
        }
        if (rank < K_N) orow[rank] = e;
    }
}

// ---------------------------------------------------------------------------
// Stage 3: gather + sigmoid(max) weight + 1x1 convs + LeakyReLU + modulation.
// One 256-thread block per (b, c_lr); indices/weights staged in LDS;
// float4 pixels -> b128 coalesced loads; gather traffic stays in L2
// (4 MB HR per batch << 192 MB L2).
// ---------------------------------------------------------------------------
__global__ __launch_bounds__(256)
void fusion_kernel(const float* __restrict__ HR, const float* __restrict__ LR,
                   const int* __restrict__ idx,
                   const float* __restrict__ w1, const float* __restrict__ b1,
                   const float* __restrict__ w2, const float* __restrict__ b2,
                   float* __restrict__ out) {
    __shared__ int   sidx[K_N];
    __shared__ float sw[K_N + 1];
    __shared__ float sconst[3];   // b1, w2, b2

    const int t  = threadIdx.x;
    const int bc = blockIdx.x;          // [0, B*C)
    const int b  = bc >> 6;             // /C_N

    if (t < K_N)                     sidx[t] = idx[(size_t)bc * K_N + t];
    if (t >= 32 && t < 32 + K_N + 1) sw[t - 32] = w1[t - 32];
    if (t == 80) sconst[0] = b1[0];
    if (t == 81) sconst[1] = w2[0];
    if (t == 82) sconst[2] = b2[0];
    __syncthreads();

    const float* __restrict__ lr  = LR + (size_t)bc * HW_N;
    const float* __restrict__ hrb = HR + (size_t)b * C_N * HW_N;
    float* __restrict__ o = out + (size_t)bc * HW_N;

    const float bias1 = sconst[0], ww2 = sconst[1], bias2 = sconst[2];
    const float w0 = sw[0];

    for (int p = t * 4; p < HW_N; p += 256 * 4) {
        const float4 l = *(const float4*)(lr + p);
        float4 acc = make_float4(w0 * l.x, w0 * l.y, w0 * l.z, w0 * l.w);
        float4 m = make_float4(-INFINITY, -INFINITY, -INFINITY, -INFINITY);
#pragma unroll 8
        for (int k = 0; k < K_N; ++k) {
            const float4 v = *(const float4*)(hrb + (size_t)sidx[k] * HW_N + p);
            const float wk = sw[k + 1];
            m.x = fmaxf(m.x, v.x); acc.x = fmaf(wk, v.x, acc.x);
            m.y = fmaxf(m.y, v.y); acc.y = fmaf(wk, v.y, acc.y);
            m.z = fmaxf(m.z, v.z); acc.z = fmaf(wk, v.z, acc.z);
            m.w = fmaxf(m.w, v.w); acc.w = fmaf(wk, v.w, acc.w);
        }
        float4 r;
        {
            float f, sg;
            f = acc.x + bias1; f = (f >= 0.f) ? f : NEG_SLOPE * f; f = fmaf(ww2, f, bias2);
            sg = 1.0f / (1.0f + __expf(-m.x)); r.x = f * (1.0f + sg);
            f = acc.y + bias1; f = (f >= 0.f) ? f : NEG_SLOPE * f; f = fmaf(ww2, f, bias2);
            sg = 1.0f / (1.0f + __expf(-m.y)); r.y = f * (1.0f + sg);
            f = acc.z + bias1; f = (f >= 0.f) ? f : NEG_SLOPE * f; f = fmaf(ww2, f, bias2);
            sg = 1.0f / (1.0f + __expf(-m.z)); r.z = f * (1.0f + sg);
            f = acc.w + bias1; f = (f >= 0.f) ? f : NEG_SLOPE * f; f = fmaf(ww2, f, bias2);
            sg = 1.0f / (1.0f + __expf(-m.w)); r.w = f * (1.0f + sg);
        }
        *(float4*)(o + p) = r;
    }
}

// ---------------------------------------------------------------------------
extern "C" void kernel_launch(void* const* d_in, const int* in_sizes, int n_in,
                              void* d_out, int out_size, void* d_ws, size_t ws_size,
                              hipStream_t stream) {
    const float* HR = (const float*)d_in[0];
    const float* LR = (const float*)d_in[1];
    const float* w1 = (const float*)d_in[2];
    const float* b1 = (const float*)d_in[3];
    const float* w2 = (const float*)d_in[4];
    const float* b2 = (const float*)d_in[5];
    float* out = (float*)d_out;

    // ws layout: partial[NSPLIT][B][16][256] f32 (1 MB), then idx[B*C][K] (64 KB)
    float* partial = (float*)d_ws;
    int*   idx     = (int*)((char*)d_ws +
                            (size_t)NSPLIT * B_N * 16 * 256 * sizeof(float));

    // Stage 1: 8 n-splits x 8 batches, 16 waves (tiles) per workgroup.
    corr_stage_kernel<<<dim3(NSPLIT, B_N), 512, 0, stream>>>(LR, HR, partial);
    // Stage 2: reduce partials + stable top-32 per (b, c_lr) row.
    topk_kernel<<<B_N * C_N, 32, 0, stream>>>(partial, idx);
    // Stage 3: gather/fuse, one block per (b, c_lr).
    fusion_kernel<<<B_N * C_N, 256, 0, stream>>>(HR, LR, idx, w1, b1, w2, b2, out);
}